// MI_70093866271368
// MI455X (gfx1250) — compile-verified
//
#include <hip/hip_runtime.h>

typedef float v2f __attribute__((ext_vector_type(2)));
typedef float v4f __attribute__((ext_vector_type(4)));
typedef float v8f __attribute__((ext_vector_type(8)));

#define N_ROWS    8192
#define N_COLS    8192
#define TPB       256
#define ROWS_PB   64
#define COLS_PB   1024   // 256 threads * 4 floats
#define ROW_BATCH 16
#define KV2       256    // 256 v2f pair-partials per row (one per thread)
#define KPAD      257    // padded v2f stride (514 floats) -> conflict-free LDS

__device__ __forceinline__ float waveReduceAdd(float v) {
    v += __shfl_xor(v, 16, 32);
    v += __shfl_xor(v, 8, 32);
    v += __shfl_xor(v, 4, 32);
    v += __shfl_xor(v, 2, 32);
    v += __shfl_xor(v, 1, 32);
    return v;
}

// Pass 1: single sweep of the 256MB table.
//  - S1  += sum(c * log2(c))   (c==0 contributes exactly 0 via the fmax trick)
//  - mx[i] += row sums         (pair-partials -> LDS -> WMMA f32 16x16x4 chain)
//  - my[j] += col sums         (per-thread register accumulation, 4 fixed cols)
__global__ __launch_bounds__(TPB) void mi_pass1(const float* __restrict__ Cmat,
                                                double* __restrict__ S1,
                                                float* __restrict__ mx,
                                                float* __restrict__ my)
{
    const int tid  = threadIdx.x;
    const int lane = tid & 31;
    const int wv   = tid >> 5;                       // 0..7 (wave32)
    const int col0 = blockIdx.x * COLS_PB + tid * 4;
    const int row0 = blockIdx.y * ROWS_PB;

    __shared__ v2f   rowPart[ROW_BATCH][KPAD];       // 16 rows x 512 K (pairs)
    __shared__ float clogPart[8];

    v4f   colAcc  = {0.f, 0.f, 0.f, 0.f};
    float clogAcc = 0.f;

    const v4f* __restrict__ basePtr =
        reinterpret_cast<const v4f*>(Cmat + (size_t)row0 * N_COLS + col0);
    const size_t rowStrideV4 = N_COLS / 4;

    for (int rb = 0; rb < ROWS_PB; rb += ROW_BATCH) {
        for (int rr = 0; rr < ROW_BATCH; ++rr) {
            const int r = rb + rr;
            // streaming, single-use data: non-temporal b128 load
            v4f c = __builtin_nontemporal_load(basePtr + (size_t)r * rowStrideV4);

            colAcc += c;

            // entropy term: c*log2(max(c, 2^-126)) == 0 exactly when c == 0
            clogAcc = fmaf(c.x, __log2f(fmaxf(c.x, 0x1p-126f)), clogAcc);
            clogAcc = fmaf(c.y, __log2f(fmaxf(c.y, 0x1p-126f)), clogAcc);
            clogAcc = fmaf(c.z, __log2f(fmaxf(c.z, 0x1p-126f)), clogAcc);
            clogAcc = fmaf(c.w, __log2f(fmaxf(c.w, 0x1p-126f)), clogAcc);

            // row pair-partial: one v_pk_add_f32 + one ds_store_b64
            v2f lo = {c.x, c.y};
            v2f hi2 = {c.z, c.w};
            rowPart[rr][tid] = lo + hi2;
        }
        __syncthreads();

        // Every wave WMMA-reduces its own K-chunk of the 16x512 partial matrix:
        // D = A(16x4) x ones(4x16) + C, chained over 16 K-windows (K = 64/wave).
        {
            const int m  = lane & 15;     // A-operand row (M)
            const int hi = lane >> 4;     // ISA A-layout: lanes 16-31 hold K=2,3
            const int kbase = wv * 32;    // this wave's v2f K-chunk
            v2f bones = {1.f, 1.f};
            v8f d = {};
#if defined(__gfx1250__) && __has_builtin(__builtin_amdgcn_wmma_f32_16x16x4_f32)
            #pragma unroll
            for (int j = 0; j < 16; ++j) {
                v2f a = rowPart[m][kbase + 2 * j + hi];
                d = __builtin_amdgcn_wmma_f32_16x16x4_f32(false, a, false, bones,
                                                          (short)0, d, false, false);
            }
#else
            // fallback with identical D lane layout (lane<16: M=q, lane>=16: M=q+8)
            (void)bones; (void)m;
            #pragma unroll
            for (int q = 0; q < 8; ++q) {
                v2f s = {0.f, 0.f};
                for (int jj = 0; jj < 32; ++jj) s += rowPart[hi * 8 + q][kbase + jj];
                d[q] = s.x + s.y;
            }
#endif
            // D[m][n] = this wave's partial row sum, replicated over n.
            // VGPR q: lanes 0-15 -> M=q, lanes 16-31 -> M=q+8.
            if (lane == 0) {
                #pragma unroll
                for (int q = 0; q < 8; ++q)
                    unsafeAtomicAdd(&mx[row0 + rb + q], d[q]);
            } else if (lane == 16) {
                #pragma unroll
                for (int q = 0; q < 8; ++q)
                    unsafeAtomicAdd(&mx[row0 + rb + 8 + q], d[q]);
            }
        }
        __syncthreads();
    }

    // column marginal: one atomic per owned column per block
    unsafeAtomicAdd(&my[col0 + 0], colAcc.x);
    unsafeAtomicAdd(&my[col0 + 1], colAcc.y);
    unsafeAtomicAdd(&my[col0 + 2], colAcc.z);
    unsafeAtomicAdd(&my[col0 + 3], colAcc.w);

    // entropy partial: wave reduce -> LDS -> one f64 atomic per block (cold path)
    float cw = waveReduceAdd(clogAcc);
    if (lane == 0) clogPart[wv] = cw;
    __syncthreads();
    if (tid == 0) {
        double s = 0.0;
        #pragma unroll
        for (int w = 0; w < 8; ++w) s += (double)clogPart[w];
        unsafeAtomicAdd(S1, s);
    }
}

// Pass 2 (tiny): MI = S1 - sum(mx*log2(mx)) - sum(my*log2(my))
__global__ __launch_bounds__(TPB) void mi_pass2(const double* __restrict__ S1,
                                                const float* __restrict__ mx,
                                                const float* __restrict__ my,
                                                float* __restrict__ out)
{
    const int tid = threadIdx.x;
    __shared__ double wsum[8];
    double acc = 0.0;
    for (int i = tid; i < N_ROWS + N_COLS; i += TPB) {
        float v = (i < N_ROWS) ? mx[i] : my[i - N_ROWS];
        if (v > 0.f) acc += (double)v * log2((double)v);
    }
    acc += __shfl_xor(acc, 16, 32);
    acc += __shfl_xor(acc, 8, 32);
    acc += __shfl_xor(acc, 4, 32);
    acc += __shfl_xor(acc, 2, 32);
    acc += __shfl_xor(acc, 1, 32);
    if ((tid & 31) == 0) wsum[tid >> 5] = acc;
    __syncthreads();
    if (tid == 0) {
        double tot = 0.0;
        #pragma unroll
        for (int w = 0; w < 8; ++w) tot += wsum[w];
        out[0] = (float)(S1[0] - tot);
    }
}

extern "C" void kernel_launch(void* const* d_in, const int* in_sizes, int n_in,
                              void* d_out, int out_size, void* d_ws, size_t ws_size,
                              hipStream_t stream) {
    (void)in_sizes; (void)n_in; (void)out_size; (void)ws_size;
    const float* Cmat = (const float*)d_in[0];
    float* out = (float*)d_out;

    // workspace: [0..7] f64 S1 accumulator, pad to 16, mx[8192] f32, my[8192] f32
    double* S1 = (double*)d_ws;
    float*  mx = (float*)((char*)d_ws + 16);
    float*  my = mx + N_ROWS;

    hipMemsetAsync(d_ws, 0, 16 + (size_t)(N_ROWS + N_COLS) * sizeof(float), stream);

    dim3 grid(N_COLS / COLS_PB, N_ROWS / ROWS_PB);  // (8, 128)
    mi_pass1<<<grid, TPB, 0, stream>>>(Cmat, S1, mx, my);
    mi_pass2<<<1, TPB, 0, stream>>>(S1, mx, my, out);
}